// KVCacheAttnBlock_68736656605950
// MI455X (gfx1250) — compile-verified
//
#include <hip/hip_runtime.h>
#include <hip/hip_bf16.h>

// Problem constants (from reference): B=8, S=1, L=4096, D=2048, H=16, HD=128
#define BB   8
#define LL   4096
#define DD   2048
#define HH   16
#define HDIM 128
#define NKEYS (LL + 1)                  // cached keys + 1 new key
#define HALF  (HDIM / 2)
#define SPLITS 8                        // split-K partitions of the key axis
#define CHUNK  513                      // ceil(NKEYS / SPLITS)

typedef __attribute__((ext_vector_type(2))) float v2f;
typedef __attribute__((ext_vector_type(8))) float v8f;

// ---------------------------------------------------------------------------
// Fp32 WMMA 16x16 tile GEMM:  out(16 x 16cols) = A(16 x D) * W(D x 16cols)
// A rows 8..15 are zero padding (real M = 8 batch rows).
// V_WMMA_F32_16X16X4_F32 layouts (ISA 7.12.2):
//   A 16x4 : lane m=lane&15; VGPR0/1 = K = (lane<16 ? 0,1 : 2,3)
//   B 4x16 : lane n=lane&15; VGPR0/1 = K rows (lane<16 ? 0,1 : 2,3)
//   C/D    : VGPR r, lanes 0-15 -> row r, lanes 16-31 -> row r+8
// ---------------------------------------------------------------------------
__device__ __forceinline__ void gemm16_tile(const float* __restrict__ A,
                                            const float* __restrict__ W,
                                            float* __restrict__ out,
                                            int n0, int lane)
{
    const int m     = lane & 15;
    const int khalf = (lane >> 4) << 1;      // 0 for lanes 0-15, 2 for 16-31
    const int ncol  = n0 + (lane & 15);

    v8f acc = {};
#pragma unroll 8
    for (int k0 = 0; k0 < DD; k0 += 4) {
        v2f a;
        if (m < BB) {
            const float2 t = *(const float2*)(A + m * DD + k0 + khalf);
            a[0] = t.x; a[1] = t.y;
        } else {
            a[0] = 0.0f; a[1] = 0.0f;
        }
        v2f b;
        b[0] = W[(k0 + khalf)     * DD + ncol];
        b[1] = W[(k0 + khalf + 1) * DD + ncol];
        // D = A*B + C   (8-arg form: neg_a, A, neg_b, B, c_mod, C, reuseA, reuseB)
        acc = __builtin_amdgcn_wmma_f32_16x16x4_f32(
            false, a, false, b, (short)0, acc, false, false);
    }
    // Store real rows 0..7 (held by lanes 0-15, VGPRs 0..7)
    if (lane < 16) {
#pragma unroll
        for (int r = 0; r < BB; ++r)
            out[r * DD + n0 + lane] = acc[r];
    }
}

// grid: 3*(D/16) single-wave blocks; block = 32 threads
__global__ void qkv_wmma_kernel(const float* __restrict__ x,
                                const float* __restrict__ Wq,
                                const float* __restrict__ Wk,
                                const float* __restrict__ Wv,
                                float* __restrict__ q,
                                float* __restrict__ kn,
                                float* __restrict__ vn)
{
    const int tiles_per_mat = DD / 16;
    const int mat = blockIdx.x / tiles_per_mat;          // 0=q 1=k 2=v
    const int n0  = (blockIdx.x % tiles_per_mat) * 16;
    const float* W = (mat == 0) ? Wq : (mat == 1) ? Wk : Wv;
    float* out     = (mat == 0) ? q  : (mat == 1) ? kn : vn;
    gemm16_tile(x, W, out, n0, (int)threadIdx.x);
}

// grid: D/16 single-wave blocks
__global__ void oproj_wmma_kernel(const float* __restrict__ attn_out,
                                  const float* __restrict__ Wo,
                                  float* __restrict__ y)
{
    gemm16_tile(attn_out, Wo, y, (int)blockIdx.x * 16, (int)threadIdx.x);
}

// ---------------------------------------------------------------------------
// RoPE at pos = cache_len (s == 1). q/kn layout: (B, H*HD) with hd contiguous.
// ---------------------------------------------------------------------------
__global__ void rope_kernel(float* __restrict__ q, float* __restrict__ kn,
                            const int* __restrict__ cache_len)
{
    const int idx = blockIdx.x * blockDim.x + threadIdx.x;
    if (idx >= BB * HH * HALF) return;
    const int j  = idx % HALF;
    const int bh = idx / HALF;

    const float pos = (float)(*cache_len);
    const float inv_freq = __powf(10000.0f, -(float)j / (float)HALF);
    const float ang = pos * inv_freq;
    float s, c;
    __sincosf(ang, &s, &c);

    const int base = bh * HDIM;
    const float q1 = q[base + j], q2 = q[base + j + HALF];
    q[base + j]        = q1 * c - q2 * s;
    q[base + j + HALF] = q1 * s + q2 * c;
    const float k1 = kn[base + j], k2 = kn[base + j + HALF];
    kn[base + j]        = k1 * c - k2 * s;
    kn[base + j + HALF] = k1 * s + k2 * c;
}

// ---------------------------------------------------------------------------
// Split-K flash-decode attention (stage 1): one workgroup (8 waves) per
// (b, h, key-chunk). 1024 workgroups / 8192 waves keep the full MI455X busy
// streaming the 512 MB KV cache at HBM rate. Each wave reads one key row per
// step: 32 lanes x float4 = 512B fully-coalesced. Produces per-chunk
// (local_max, local_expsum, unnormalized weighted-V accumulator).
// ---------------------------------------------------------------------------
__global__ void attn_partial_kernel(const float* __restrict__ q,
                                    const float* __restrict__ kn,
                                    const float* __restrict__ vn,
                                    const float* __restrict__ k_cache,
                                    const float* __restrict__ v_cache,
                                    float* __restrict__ pm,   // [B*H*SPLITS]
                                    float* __restrict__ ps,   // [B*H*SPLITS]
                                    float* __restrict__ pacc) // [B*H*SPLITS][HDIM]
{
    __shared__ float s_scores[CHUNK];
    __shared__ float s_red[8];
    __shared__ float s_bcast[2];
    __shared__ float s_part[8][HDIM];

    const int bh   = blockIdx.x / SPLITS;
    const int p    = blockIdx.x % SPLITS;
    const int b    = bh / HH;
    const int h    = bh % HH;
    const int kbeg = p * CHUNK;
    const int kend = (kbeg + CHUNK < NKEYS) ? (kbeg + CHUNK) : NKEYS;
    const int nk   = kend - kbeg;

    const int tid  = threadIdx.x;
    const int lane = tid & 31;
    const int wave = tid >> 5;
    const int NW   = 8;
    const float scale = 0.088388347648318447f;   // 1/sqrt(128)

    const float4 qv = *(const float4*)(q + b * DD + h * HDIM + lane * 4);
    const float* knew = kn + b * DD + h * HDIM;
    const float* vnew = vn + b * DD + h * HDIM;

    // ---- scores for this chunk: one key per wave per iteration -----------
    for (int key = kbeg + wave; key < kend; key += NW) {
        const float* krow = (key < LL)
            ? (k_cache + (((size_t)b * LL + key) * HH + h) * HDIM)
            : knew;
        const float4 kv = *(const float4*)(krow + lane * 4);
        float d = qv.x * kv.x + qv.y * kv.y + qv.z * kv.z + qv.w * kv.w;
#pragma unroll
        for (int off = 16; off > 0; off >>= 1)
            d += __shfl_down(d, off, 32);
        if (lane == 0) s_scores[key - kbeg] = d * scale;
    }
    __syncthreads();

    // ---- local max --------------------------------------------------------
    float mx = -3.0e38f;
    for (int i = tid; i < nk; i += 256) mx = fmaxf(mx, s_scores[i]);
#pragma unroll
    for (int off = 16; off > 0; off >>= 1)
        mx = fmaxf(mx, __shfl_down(mx, off, 32));
    if (lane == 0) s_red[wave] = mx;
    __syncthreads();
    if (tid == 0) {
        float m = s_red[0];
#pragma unroll
        for (int w = 1; w < 8; ++w) m = fmaxf(m, s_red[w]);
        s_bcast[0] = m;
    }
    __syncthreads();
    mx = s_bcast[0];

    // ---- exp + local sum --------------------------------------------------
    float sum = 0.0f;
    for (int i = tid; i < nk; i += 256) {
        const float e = __expf(s_scores[i] - mx);
        s_scores[i] = e;
        sum += e;
    }
#pragma unroll
    for (int off = 16; off > 0; off >>= 1)
        sum += __shfl_down(sum, off, 32);
    __syncthreads();                 // s_red reuse safety
    if (lane == 0) s_red[wave] = sum;
    __syncthreads();
    if (tid == 0) {
        float t = 0.0f;
#pragma unroll
        for (int w = 0; w < 8; ++w) t += s_red[w];
        pm[blockIdx.x] = mx;
        ps[blockIdx.x] = t;
    }

    // ---- unnormalized weighted-V accumulation -----------------------------
    float4 acc = make_float4(0.f, 0.f, 0.f, 0.f);
    for (int key = kbeg + wave; key < kend; key += NW) {
        const float* vrow = (key < LL)
            ? (v_cache + (((size_t)b * LL + key) * HH + h) * HDIM)
            : vnew;
        const float4 vv = *(const float4*)(vrow + lane * 4);
        const float w = s_scores[key - kbeg];
        acc.x += w * vv.x; acc.y += w * vv.y;
        acc.z += w * vv.z; acc.w += w * vv.w;
    }
    *(float4*)&s_part[wave][lane * 4] = acc;
    __syncthreads();

    if (tid < HDIM) {
        float r = 0.0f;
#pragma unroll
        for (int w = 0; w < 8; ++w) r += s_part[w][tid];
        pacc[(size_t)blockIdx.x * HDIM + tid] = r;
    }
}

// ---------------------------------------------------------------------------
// Split-K combine (stage 2): log-sum-exp merge of the SPLITS partials.
// grid: B*H blocks, HDIM threads. attn output layout: bh*HDIM == b*D + h*HD.
// ---------------------------------------------------------------------------
__global__ void attn_combine_kernel(const float* __restrict__ pm,
                                    const float* __restrict__ ps,
                                    const float* __restrict__ pacc,
                                    float* __restrict__ out)
{
    __shared__ float sm[SPLITS];
    __shared__ float ss[SPLITS];
    const int bh  = blockIdx.x;
    const int tid = threadIdx.x;

    if (tid < SPLITS) {
        sm[tid] = pm[bh * SPLITS + tid];
        ss[tid] = ps[bh * SPLITS + tid];
    }
    __syncthreads();

    float M = -3.0e38f;
#pragma unroll
    for (int p = 0; p < SPLITS; ++p) M = fmaxf(M, sm[p]);

    float total = 0.0f;
    float r = 0.0f;
#pragma unroll
    for (int p = 0; p < SPLITS; ++p) {
        const float w = __expf(sm[p] - M);
        total += ss[p] * w;
        r += w * pacc[(size_t)(bh * SPLITS + p) * HDIM + tid];
    }
    out[bh * HDIM + tid] = r / total;
}

// ---------------------------------------------------------------------------
extern "C" void kernel_launch(void* const* d_in, const int* in_sizes, int n_in,
                              void* d_out, int out_size, void* d_ws, size_t ws_size,
                              hipStream_t stream) {
    const float* x        = (const float*)d_in[0];
    const float* k_cache  = (const float*)d_in[1];
    const float* v_cache  = (const float*)d_in[2];
    const float* Wq       = (const float*)d_in[3];
    const float* Wk       = (const float*)d_in[4];
    const float* Wv       = (const float*)d_in[5];
    const float* Wo       = (const float*)d_in[6];
    const int*   cache_len= (const int*)d_in[7];

    float* ws   = (float*)d_ws;
    float* q    = ws;                        // B*D
    float* kn   = ws + BB * DD;              // B*D
    float* vn   = ws + 2 * BB * DD;          // B*D
    float* aout = ws + 3 * BB * DD;          // B*D
    float* pm   = ws + 4 * BB * DD;          // B*H*SPLITS
    float* ps   = pm + BB * HH * SPLITS;     // B*H*SPLITS
    float* pacc = ps + BB * HH * SPLITS;     // B*H*SPLITS*HDIM

    // 1) QKV projections (fp32 WMMA, one pass over each weight matrix)
    qkv_wmma_kernel<<<3 * (DD / 16), 32, 0, stream>>>(x, Wq, Wk, Wv, q, kn, vn);

    // 2) RoPE at pos = cache_len
    rope_kernel<<<(BB * HH * HALF + 255) / 256, 256, 0, stream>>>(q, kn, cache_len);

    // 3) Split-K flash decode: 1024 workgroups stream the 512MB KV cache once
    attn_partial_kernel<<<BB * HH * SPLITS, 256, 0, stream>>>(
        q, kn, vn, k_cache, v_cache, pm, ps, pacc);
    attn_combine_kernel<<<BB * HH, HDIM, 0, stream>>>(pm, ps, pacc, aout);

    // 4) Output projection (fp32 WMMA)
    oproj_wmma_kernel<<<DD / 16, 32, 0, stream>>>(aout, Wo, (float*)d_out);
}